// Net_75831942578615
// MI455X (gfx1250) — compile-verified
//
#include <hip/hip_runtime.h>
#include <math.h>

#define KS 5
#define KBINS 125

typedef __attribute__((ext_vector_type(2))) float v2f;
typedef __attribute__((ext_vector_type(8))) float v8f;

// ---------- small utilities ----------

__device__ __forceinline__ void atomicMaxF(float* addr, float val) {
    // order-preserving float atomic max (handles mixed signs; addr init to -inf ok)
    if (val >= 0.0f) atomicMax((int*)addr, __float_as_int(val));
    else             atomicMin((unsigned int*)addr, __float_as_uint(val));
}

__global__ void fill_f32(float* __restrict__ p, float v, size_t n) {
    size_t i = (size_t)blockIdx.x * blockDim.x + threadIdx.x;
    if (i < n) p[i] = v;
}

__global__ void valid_init(const int* __restrict__ row, const int* __restrict__ col,
                           float* __restrict__ valid, int E) {
    int e = blockIdx.x * blockDim.x + threadIdx.x;
    if (e < E) valid[e] = (row[e] != col[e]) ? 1.0f : 0.0f;
}

// ---------- pad/pack: X[n,cin] -> Xp[n,cinP];  W[nb,cin,cout] -> Wp[nb,cinP,coutP] ----------

__global__ void pack_x(const float* __restrict__ X, float* __restrict__ Xp,
                       int n, int cin, int cinP) {
    size_t idx = (size_t)blockIdx.x * blockDim.x + threadIdx.x;
    size_t tot = (size_t)n * cinP;
    if (idx >= tot) return;
    int r = (int)(idx / cinP), i = (int)(idx % cinP);
    Xp[idx] = (i < cin) ? X[(size_t)r * cin + i] : 0.0f;
}

__global__ void pack_w(const float* __restrict__ W, float* __restrict__ Wp,
                       int nb, int cin, int cout, int cinP, int coutP) {
    size_t idx = (size_t)blockIdx.x * blockDim.x + threadIdx.x;
    size_t tot = (size_t)nb * cinP * coutP;
    if (idx >= tot) return;
    int b   = (int)(idx / ((size_t)cinP * coutP));
    int rem = (int)(idx % ((size_t)cinP * coutP));
    int i = rem / coutP, o = rem % coutP;
    Wp[idx] = (i < cin && o < cout) ? W[((size_t)b * cin + i) * cout + o] : 0.0f;
}

// ---------- global max |pos[col]-pos[row]| over valid edges ----------

__global__ void edge_maxabs(const float* __restrict__ pos, const int* __restrict__ row,
                            const int* __restrict__ col, const float* __restrict__ valid,
                            float* __restrict__ maxd, int E) {
    __shared__ float red[256];
    float m = 0.0f;
    for (int e = blockIdx.x * blockDim.x + threadIdx.x; e < E; e += gridDim.x * blockDim.x) {
        if (valid[e] != 0.0f) {
            int r = row[e], c = col[e];
            #pragma unroll
            for (int d = 0; d < 3; ++d)
                m = fmaxf(m, fabsf(pos[c * 3 + d] - pos[r * 3 + d]));
        }
    }
    red[threadIdx.x] = m;
    __syncthreads();
    for (int s = blockDim.x / 2; s > 0; s >>= 1) {
        if ((int)threadIdx.x < s) red[threadIdx.x] = fmaxf(red[threadIdx.x], red[threadIdx.x + s]);
        __syncthreads();
    }
    if (threadIdx.x == 0)
        atomicMax((unsigned int*)maxd, __float_as_uint(red[0])); // all values >= 0
}

// ---------- pseudo-coordinate / kernel-bin helper ----------

__device__ __forceinline__ void edge_pseudo(const float* __restrict__ pos, int r, int c,
                                            float inv2m, float* u, int* kout) {
    int b[3];
    #pragma unroll
    for (int d = 0; d < 3; ++d) {
        float ud = (pos[c * 3 + d] - pos[r * 3 + d]) * inv2m + 0.5f;
        ud = fminf(fmaxf(ud, 0.0f), 1.0f);
        u[d] = ud;
        int bi = (int)floorf(ud * (float)KS);
        b[d] = min(max(bi, 0), KS - 1);
    }
    *kout = (b[0] * KS + b[1]) * KS + b[2];
}

// ---------- fp32 WMMA batched GEMM on padded operands (fully branchless) ----------
// Y[(m*nb + b)*coutP + o] = sum_i Xp[m,i] * Wp[b,i,o]
// One wave per 16x16 tile; CINP compile-time => full unroll, loads hoisted,
// CINP/4 chained V_WMMA_F32_16X16X4_F32. n is a multiple of 16 (no row guards).

template<int CINP>
__global__ void wmma_gemm_packed(const float* __restrict__ Xp, const float* __restrict__ Wp,
                                 float* __restrict__ Y, int coutP, int nb) {
    int mt = blockIdx.x * 16;
    int nt = blockIdx.y * 16;
    int b  = blockIdx.z;
    int lane = threadIdx.x;      // 0..31
    int half = lane >> 4;        // 0 or 1
    int l16  = lane & 15;

    // A frag: 16x4 f32: lanes 0-15 hold K=i0,i0+1 of row M=l16; lanes 16-31 hold K=i0+2,i0+3
    const float* arow = Xp + (size_t)(mt + l16) * CINP + half * 2;
    // B frag: 4x16 f32: lanes 0-15 hold K=i0,i0+1 of col N=l16; lanes 16-31 hold K=i0+2,i0+3
    const float* bcol = Wp + (size_t)b * CINP * coutP + (size_t)(half * 2) * coutP + nt + l16;

    v8f acc = {};
    #pragma unroll
    for (int i0 = 0; i0 < CINP; i0 += 4) {
        v2f a = *(const v2f*)(arow + i0);                 // 8B-aligned -> b64 load
        v2f bf;
        bf.x = bcol[(size_t)i0 * coutP];
        bf.y = bcol[(size_t)i0 * coutP + coutP];
        acc = __builtin_amdgcn_wmma_f32_16x16x4_f32(false, a, false, bf,
                                                    (short)0, acc, false, false);
    }

    // D: VGPR r holds M = mt+r (lanes 0-15) / M = mt+r+8 (lanes 16-31), N = nt + lane%16
    #pragma unroll
    for (int r = 0; r < 8; ++r) {
        int row = mt + r + half * 8;
        Y[((size_t)row * nb + b) * coutP + nt + l16] = acc[r];
    }
}

// ---------- edge message pass, per-edge matvec (levels 1-3) ----------

template<int CIN, int COUT>
__global__ void edge_conv_mat(const float* __restrict__ X, const float* __restrict__ pos,
                              const int* __restrict__ row, const int* __restrict__ col,
                              const float* __restrict__ valid, const float* __restrict__ w,
                              const float* __restrict__ wg, const float* __restrict__ bg,
                              const float* __restrict__ maxd,
                              float* __restrict__ agg, float* __restrict__ invagg,
                              float* __restrict__ deg, int E) {
    int e = blockIdx.x * blockDim.x + threadIdx.x;
    if (e >= E) return;
    float v = valid[e];
    if (v == 0.0f) return;
    int r = row[e], c = col[e];
    float inv2m = 0.5f / fmaxf(maxd[0], 1e-12f);
    float u[3]; int k;
    edge_pseudo(pos, r, c, inv2m, u, &k);

    const float* wk = w + (size_t)k * CIN * COUT;
    float xv[CIN];
    #pragma unroll
    for (int i = 0; i < CIN; ++i) xv[i] = X[(size_t)r * CIN + i];
    #pragma unroll
    for (int o = 0; o < COUT; ++o) {
        float s = 0.0f;
        #pragma unroll
        for (int i = 0; i < CIN; ++i) s += xv[i] * wk[i * COUT + o];
        atomicAdd(&agg[(size_t)c * COUT + o], s);
    }
    #pragma unroll
    for (int j = 0; j < 8; ++j) {
        float g = bg[j] + u[0] * wg[0 * 8 + j] + u[1] * wg[1 * 8 + j] + u[2] * wg[2 * 8 + j];
        atomicAdd(&invagg[(size_t)c * 8 + j], g);
    }
    atomicAdd(&deg[c], v);
}

// ---------- edge message pass, gather of precomputed Y (levels 4-5) ----------

__global__ void edge_conv_gather(const float* __restrict__ Y, const float* __restrict__ pos,
                                 const int* __restrict__ row, const int* __restrict__ col,
                                 const float* __restrict__ valid, const float* __restrict__ wg,
                                 const float* __restrict__ bg, const float* __restrict__ maxd,
                                 float* __restrict__ agg, float* __restrict__ invagg,
                                 float* __restrict__ deg, int E, int cout, int coutP) {
    int e = blockIdx.x * blockDim.x + threadIdx.x;
    if (e >= E) return;
    float v = valid[e];
    if (v == 0.0f) return;
    int r = row[e], c = col[e];
    float inv2m = 0.5f / fmaxf(maxd[0], 1e-12f);
    float u[3]; int k;
    edge_pseudo(pos, r, c, inv2m, u, &k);

    const float* yr = Y + ((size_t)r * KBINS + k) * coutP;
    for (int o = 0; o < cout; ++o)
        atomicAdd(&agg[(size_t)c * cout + o], yr[o]);
    #pragma unroll
    for (int j = 0; j < 8; ++j) {
        float g = bg[j] + u[0] * wg[0 * 8 + j] + u[1] * wg[1 * 8 + j] + u[2] * wg[2 * 8 + j];
        atomicAdd(&invagg[(size_t)c * 8 + j], g);
    }
    atomicAdd(&deg[c], v);
}

// ---------- conv finalize: main = agg/degc + x@wr + b ; inv = invagg/degc ; log1p(deg) ----------

__global__ void conv_finalize(const float* __restrict__ agg, const float* __restrict__ invagg,
                              const float* __restrict__ deg, const float* __restrict__ xr,
                              const float* __restrict__ bias, float* __restrict__ hout,
                              int n, int cout, int coutP) {
    int i = blockIdx.x * blockDim.x + threadIdx.x;
    if (i >= n) return;
    float d  = deg[i];
    float dc = fmaxf(d, 1.0f);
    int co = cout + 9;
    for (int o = 0; o < cout; ++o)
        hout[(size_t)i * co + o] = agg[(size_t)i * cout + o] / dc
                                 + xr[(size_t)i * coutP + o] + bias[o];
    #pragma unroll
    for (int j = 0; j < 8; ++j)
        hout[(size_t)i * co + cout + j] = invagg[(size_t)i * 8 + j] / dc;
    hout[(size_t)i * co + cout + 8] = log1pf(d);
}

// ---------- voxel max pool ----------

__global__ void pool_scatter(const float* __restrict__ h, const float* __restrict__ pos,
                             const float* __restrict__ nmask, int* __restrict__ cluster,
                             float* __restrict__ px, float* __restrict__ cnt,
                             float* __restrict__ ppos, int n, int ch, int grid,
                             float start, float invsize) {
    int i = blockIdx.x * blockDim.x + threadIdx.x;
    if (i >= n) return;
    int c3[3];
    #pragma unroll
    for (int d = 0; d < 3; ++d) {
        int b = (int)floorf((pos[i * 3 + d] - start) * invsize);
        c3[d] = min(max(b, 0), grid - 1);
    }
    int cl = (c3[0] * grid + c3[1]) * grid + c3[2];
    cluster[i] = cl;
    float nm = nmask ? nmask[i] : 1.0f;
    if (nm > 0.0f) {
        for (int c = 0; c < ch; ++c)
            atomicMaxF(&px[(size_t)cl * ch + c], h[(size_t)i * ch + c]);
        atomicAdd(&cnt[cl], nm);
        #pragma unroll
        for (int d = 0; d < 3; ++d)
            atomicAdd(&ppos[cl * 3 + d], pos[i * 3 + d] * nm);
    }
}

__global__ void pool_finalize(float* __restrict__ px, const float* __restrict__ cnt,
                              const float* __restrict__ ppos, float* __restrict__ posout,
                              float* __restrict__ occ, int v, int ch) {
    int i = blockIdx.x * blockDim.x + threadIdx.x;
    if (i >= v) return;
    float c = cnt[i];
    occ[i] = (c > 0.0f) ? 1.0f : 0.0f;
    if (c <= 0.0f)
        for (int j = 0; j < ch; ++j) px[(size_t)i * ch + j] = 0.0f;
    float dc = fmaxf(c, 1.0f);
    #pragma unroll
    for (int d = 0; d < 3; ++d) posout[i * 3 + d] = ppos[i * 3 + d] / dc;
}

__global__ void edge_remap(const int* __restrict__ row, const int* __restrict__ col,
                           const float* __restrict__ valid, const int* __restrict__ cluster,
                           int* __restrict__ nrow, int* __restrict__ ncol,
                           float* __restrict__ nvalid, int E) {
    int e = blockIdx.x * blockDim.x + threadIdx.x;
    if (e >= E) return;
    int nr = cluster[row[e]], nc = cluster[col[e]];
    nrow[e] = nr; ncol[e] = nc;
    nvalid[e] = valid[e] * ((nr != nc) ? 1.0f : 0.0f);
}

// ---------- final FC + log_softmax ----------

__global__ void fc_logsoftmax(const float* __restrict__ z, const float* __restrict__ fcw,
                              const float* __restrict__ fcb, float* __restrict__ out) {
    __shared__ float logits[10];
    int t = threadIdx.x;
    if (t < 10) {
        float s = fcb[t];
        for (int i = 0; i < 392; ++i) s += z[i] * fcw[i * 10 + t];
        logits[t] = s;
    }
    __syncthreads();
    if (t == 0) {
        float mx = logits[0];
        for (int o = 1; o < 10; ++o) mx = fmaxf(mx, logits[o]);
        float se = 0.0f;
        for (int o = 0; o < 10; ++o) se += expf(logits[o] - mx);
        float lse = mx + logf(se);
        for (int o = 0; o < 10; ++o) out[o] = logits[o] - lse;
    }
}

// ---------- host orchestration ----------

extern "C" void kernel_launch(void* const* d_in, const int* in_sizes, int n_in,
                              void* d_out, int out_size, void* d_ws, size_t ws_size,
                              hipStream_t stream) {
    const float* x0   = (const float*)d_in[0];
    const float* pos0 = (const float*)d_in[1];
    const int*   row0 = (const int*)d_in[2];
    const int*   col0 = (const int*)d_in[3];
    const int N = in_sizes[0];   // x is [N,1]
    const int E = in_sizes[2];

    const float *W[5], *WR[5], *B[5], *WG[5], *BG[5];
    for (int l = 0; l < 5; ++l) {
        W[l]  = (const float*)d_in[4 + 5 * l + 0];
        WR[l] = (const float*)d_in[4 + 5 * l + 1];
        B[l]  = (const float*)d_in[4 + 5 * l + 2];
        WG[l] = (const float*)d_in[4 + 5 * l + 3];
        BG[l] = (const float*)d_in[4 + 5 * l + 4];
    }
    const float* fcw = (const float*)d_in[29];
    const float* fcb = (const float*)d_in[30];

    // workspace carve-up (all L2-resident: ~85 MB total on a 192 MB L2)
    char* wsb = (char*)d_ws;
    size_t off = 0;
    auto alloc = [&](size_t bytes) -> void* {
        void* p = wsb + off;
        off += (bytes + 255) & ~(size_t)255;
        return p;
    };
    float* valid0 = (float*)alloc((size_t)E * 4);
    int*   rowA   = (int*)  alloc((size_t)E * 4);
    int*   colA   = (int*)  alloc((size_t)E * 4);
    float* validA = (float*)alloc((size_t)E * 4);
    int*   rowB   = (int*)  alloc((size_t)E * 4);
    int*   colB   = (int*)  alloc((size_t)E * 4);
    float* validB = (float*)alloc((size_t)E * 4);
    float* HA     = (float*)alloc((size_t)N * 17 * 4);      // conv outputs (max N x 17)
    float* HB     = (float*)alloc((size_t)32768 * 17 * 4);  // pool outputs (max 32768 x 17)
    float* posA   = (float*)alloc((size_t)32768 * 3 * 4);
    float* posB   = (float*)alloc((size_t)4096 * 3 * 4);
    float* nmA    = (float*)alloc((size_t)32768 * 4);
    float* nmB    = (float*)alloc((size_t)4096 * 4);
    int*   clus   = (int*)  alloc((size_t)N * 4);
    float* agg    = (float*)alloc((size_t)N * 8 * 4);       // max n*cout = 100000*8
    float* invagg = (float*)alloc((size_t)N * 8 * 4);
    float* xr     = (float*)alloc((size_t)N * 16 * 4);      // max n*coutP = 100000*16
    float* deg    = (float*)alloc((size_t)N * 4);
    float* maxd   = (float*)alloc(256);
    float* Ybuf   = (float*)alloc((size_t)512 * KBINS * 32 * 4); // max padded Y (level 4)
    float* Xp     = (float*)alloc((size_t)32768 * 20 * 4);  // max n*cinP
    float* Wp     = (float*)alloc((size_t)KBINS * 44 * 48 * 4); // max nb*cinP*coutP (level 5)
    float* pcnt   = (float*)alloc((size_t)32768 * 4);
    float* pppos  = (float*)alloc((size_t)32768 * 3 * 4);

    auto fill = [&](float* p, float v, size_t cnt) {
        int blocks = (int)((cnt + 255) / 256);
        fill_f32<<<blocks, 256, 0, stream>>>(p, v, cnt);
    };
    const int EB = (E + 255) / 256;

    auto do_pack_x = [&](const float* X, int n, int cin, int cinP) {
        size_t tot = (size_t)n * cinP;
        pack_x<<<(int)((tot + 255) / 256), 256, 0, stream>>>(X, Xp, n, cin, cinP);
    };
    auto do_pack_w = [&](const float* Wsrc, int nb, int cin, int cout, int cinP, int coutP) {
        size_t tot = (size_t)nb * cinP * coutP;
        pack_w<<<(int)((tot + 255) / 256), 256, 0, stream>>>(Wsrc, Wp, nb, cin, cout, cinP, coutP);
    };
    auto gemm = [&](int cinP, float* Y, int n, int coutP, int nb) {
        dim3 g(n / 16, coutP / 16, nb);
        switch (cinP) {
            case 4:  wmma_gemm_packed<4> <<<g, 32, 0, stream>>>(Xp, Wp, Y, coutP, nb); break;
            case 20: wmma_gemm_packed<20><<<g, 32, 0, stream>>>(Xp, Wp, Y, coutP, nb); break;
            case 28: wmma_gemm_packed<28><<<g, 32, 0, stream>>>(Xp, Wp, Y, coutP, nb); break;
            case 36: wmma_gemm_packed<36><<<g, 32, 0, stream>>>(Xp, Wp, Y, coutP, nb); break;
            default: wmma_gemm_packed<44><<<g, 32, 0, stream>>>(Xp, Wp, Y, coutP, nb); break;
        }
    };

    // conv prologue: maxd reduction, zero accumulators, pack X, root GEMM x@wr via WMMA
    auto conv_pre = [&](int n, int cin, int cout, const float* X, const float* P,
                        const int* R, const int* C, const float* V, const float* wr,
                        int cinP, int coutP) {
        fill(maxd, 0.0f, 1);
        edge_maxabs<<<512, 256, 0, stream>>>(P, R, C, V, maxd, E);
        fill(agg,    0.0f, (size_t)n * cout);
        fill(invagg, 0.0f, (size_t)n * 8);
        fill(deg,    0.0f, (size_t)n);
        do_pack_x(X, n, cin, cinP);
        do_pack_w(wr, 1, cin, cout, cinP, coutP);
        gemm(cinP, xr, n, coutP, 1);
    };
    auto conv_post = [&](int n, int cout, int coutP, const float* bias, float* hout) {
        conv_finalize<<<(n + 255) / 256, 256, 0, stream>>>(agg, invagg, deg, xr, bias,
                                                           hout, n, cout, coutP);
    };
    auto pool = [&](int n, int ch, int grid, float start, float invsize,
                    const float* h, const float* P, const float* nm,
                    float* hout, float* posout, float* occ) {
        int v = grid * grid * grid;
        fill(hout, -INFINITY, (size_t)v * ch);
        fill(pcnt, 0.0f, (size_t)v);
        fill(pppos, 0.0f, (size_t)v * 3);
        pool_scatter<<<(n + 255) / 256, 256, 0, stream>>>(h, P, nm, clus, hout, pcnt, pppos,
                                                          n, ch, grid, start, invsize);
        pool_finalize<<<(v + 255) / 256, 256, 0, stream>>>(hout, pcnt, pppos, posout, occ, v, ch);
    };

    // ----- Level 1: n=N, cin=1 -> 8 (+9);  cinP=4, coutP=16 -----
    valid_init<<<EB, 256, 0, stream>>>(row0, col0, valid0, E);
    conv_pre(N, 1, 8, x0, pos0, row0, col0, valid0, WR[0], 4, 16);
    edge_conv_mat<1, 8><<<EB, 256, 0, stream>>>(x0, pos0, row0, col0, valid0,
                                                W[0], WG[0], BG[0], maxd, agg, invagg, deg, E);
    conv_post(N, 8, 16, B[0], HA);
    pool(N, 17, 32, 0.0f, 32.0f, HA, pos0, nullptr, HB, posA, nmA);
    edge_remap<<<EB, 256, 0, stream>>>(row0, col0, valid0, clus, rowA, colA, validA, E);

    // ----- Level 2: n=32768, cin=17 -> 16 (+9);  cinP=20, coutP=16 -----
    conv_pre(32768, 17, 16, HB, posA, rowA, colA, validA, WR[1], 20, 16);
    edge_conv_mat<17, 16><<<EB, 256, 0, stream>>>(HB, posA, rowA, colA, validA,
                                                  W[1], WG[1], BG[1], maxd, agg, invagg, deg, E);
    conv_post(32768, 16, 16, B[1], HA);
    pool(32768, 25, 16, 0.0f, 16.0f, HA, posA, nmA, HB, posB, nmB);
    edge_remap<<<EB, 256, 0, stream>>>(rowA, colA, validA, clus, rowB, colB, validB, E);

    // ----- Level 3: n=4096, cin=25 -> 24 (+9);  cinP=28, coutP=32 -----
    conv_pre(4096, 25, 24, HB, posB, rowB, colB, validB, WR[2], 28, 32);
    edge_conv_mat<25, 24><<<EB, 256, 0, stream>>>(HB, posB, rowB, colB, validB,
                                                  W[2], WG[2], BG[2], maxd, agg, invagg, deg, E);
    conv_post(4096, 24, 32, B[2], HA);
    pool(4096, 33, 8, 0.0f, 8.0f, HA, posB, nmB, HB, posA, nmA);
    edge_remap<<<EB, 256, 0, stream>>>(rowB, colB, validB, clus, rowA, colA, validA, E);

    // ----- Level 4: n=512, cin=33 -> 32 (+9);  cinP=36, coutP=32; batched WMMA over 125 bins -----
    conv_pre(512, 33, 32, HB, posA, rowA, colA, validA, WR[3], 36, 32);
    do_pack_w(W[3], KBINS, 33, 32, 36, 32);
    gemm(36, Ybuf, 512, 32, KBINS);
    edge_conv_gather<<<EB, 256, 0, stream>>>(Ybuf, posA, rowA, colA, validA,
                                             WG[3], BG[3], maxd, agg, invagg, deg, E, 32, 32);
    conv_post(512, 32, 32, B[3], HA);
    pool(512, 41, 4, 0.0f, 4.0f, HA, posA, nmA, HB, posB, nmB);
    edge_remap<<<EB, 256, 0, stream>>>(rowA, colA, validA, clus, rowB, colB, validB, E);

    // ----- Level 5: n=64, cin=41 -> 40 (+9);  cinP=44, coutP=48; batched WMMA over 125 bins -----
    conv_pre(64, 41, 40, HB, posB, rowB, colB, validB, WR[4], 44, 48);
    do_pack_w(W[4], KBINS, 41, 40, 44, 48);
    gemm(44, Ybuf, 64, 48, KBINS);
    edge_conv_gather<<<EB, 256, 0, stream>>>(Ybuf, posB, rowB, colB, validB,
                                             WG[4], BG[4], maxd, agg, invagg, deg, E, 40, 48);
    conv_post(64, 40, 48, B[4], HA);
    pool(64, 49, 2, -0.5f, 1.0f, HA, posB, nmB, HB, posA, nmA);

    // ----- FC [1,392] @ [392,10] + log_softmax -----
    fc_logsoftmax<<<1, 32, 0, stream>>>(HB, fcw, fcb, (float*)d_out);
}